// CGPLayer_60773787239141
// MI455X (gfx1250) — compile-verified
//
#include <hip/hip_runtime.h>

// CGP elementwise DAG, 4.19M rows x 8 f32 in -> 8 f32 out.
// Streaming kernel: 256 MiB total traffic, ~11.5 us floor at 23.3 TB/s.
// gfx1250 specifics: b128 global ops with th:TH_*_NT hints (one-pass stream,
// keep L2 clean), native v_sin/v_cos/v_tanh trans ops, wave32 256-thread
// blocks, 4 rows/thread with a guard-free fast path (no exec masking, no
// zero-init) and batched loads for memory-level parallelism.

#define ROWS_PER_THREAD 4
#define BLOCK_SIZE 256

// Native clang vector type: required by __builtin_nontemporal_load/store.
typedef __attribute__((ext_vector_type(4))) float v4f;

__device__ __forceinline__ float ftanh(float x) {
#if __has_builtin(__builtin_amdgcn_tanhf)
  return __builtin_amdgcn_tanhf(x);
#else
  float r;
  asm("v_tanh_f32 %0, %1" : "=v"(r) : "v"(x));
  return r;
#endif
}

__device__ __forceinline__ void cgp_row(const v4f lo, const v4f hi,
                                        float c0, float c1, float c2, float c3,
                                        v4f& o_lo, v4f& o_hi) {
  const float x0 = lo.x, x1 = lo.y, x2 = lo.z, x3 = lo.w;
  const float x4v = hi.x, x5 = hi.y, x6 = hi.z, x7 = hi.w;

  const float n0  = x0 + x1;
  const float n1  = x2 * x3;
  const float n2  = __sinf(x4v);
  const float n3  = ftanh(x5 + c0);
  const float n4  = n0 * n1;
  const float n5  = n2 + n3;
  const float n6  = n4 - n5;
  const float n7  = __cosf(n6);
  const float n8  = n0 * c1;
  const float n9  = n7 + n8;
  const float n10 = ftanh(n9);
  const float n11 = x6 * x7;
  const float n12 = n11 + c2;
  const float n13 = __sinf(n12);
  const float n14 = n10 * n13;
  const float n15 = n14 + c3;

  // stack order: [n15, n10, n13, n9, n4, n5, n7, n12]
  o_lo = (v4f){n15, n10, n13, n9};
  o_hi = (v4f){n4, n5, n7, n12};
}

__global__ __launch_bounds__(BLOCK_SIZE) void cgp_kernel(
    const v4f* __restrict__ x4, const float* __restrict__ ephs,
    v4f* __restrict__ out4, int nrows, int total_threads) {
  // Ephemeral constants are wave-uniform -> one s_load_b128.
  const float c0 = ephs[0];
  const float c1 = ephs[1];
  const float c2 = ephs[2];
  const float c3 = ephs[3];

  const int t = blockIdx.x * BLOCK_SIZE + threadIdx.x;

  if (t + (ROWS_PER_THREAD - 1) * total_threads < nrows) {
    // ---- Fast path: all rows in range; no guards, no exec masking. ----
    v4f lo[ROWS_PER_THREAD];
    v4f hi[ROWS_PER_THREAD];

    // One clause of 8 NT b128 loads in flight per lane.
#pragma unroll
    for (int r = 0; r < ROWS_PER_THREAD; ++r) {
      const int row = t + r * total_threads;
      lo[r] = __builtin_nontemporal_load(&x4[2 * row]);
      hi[r] = __builtin_nontemporal_load(&x4[2 * row + 1]);
    }

#pragma unroll
    for (int r = 0; r < ROWS_PER_THREAD; ++r) {
      const int row = t + r * total_threads;
      v4f o_lo, o_hi;
      cgp_row(lo[r], hi[r], c0, c1, c2, c3, o_lo, o_hi);
      __builtin_nontemporal_store(o_lo, &out4[2 * row]);
      __builtin_nontemporal_store(o_hi, &out4[2 * row + 1]);
    }
  } else {
    // ---- Ragged tail (dynamically dead when nrows % (BLOCK*RPT) == 0). ----
#pragma unroll
    for (int r = 0; r < ROWS_PER_THREAD; ++r) {
      const int row = t + r * total_threads;
      if (row < nrows) {
        const v4f lo = __builtin_nontemporal_load(&x4[2 * row]);
        const v4f hi = __builtin_nontemporal_load(&x4[2 * row + 1]);
        v4f o_lo, o_hi;
        cgp_row(lo, hi, c0, c1, c2, c3, o_lo, o_hi);
        __builtin_nontemporal_store(o_lo, &out4[2 * row]);
        __builtin_nontemporal_store(o_hi, &out4[2 * row + 1]);
      }
    }
  }
}

extern "C" void kernel_launch(void* const* d_in, const int* in_sizes, int n_in,
                              void* d_out, int out_size, void* d_ws, size_t ws_size,
                              hipStream_t stream) {
  (void)n_in; (void)out_size; (void)d_ws; (void)ws_size;

  const float* x    = (const float*)d_in[0];   // [BATCH, 8] f32, row-major
  const float* ephs = (const float*)d_in[1];   // [4] f32
  float* out        = (float*)d_out;           // [BATCH, 8] f32, row-major

  const int nrows = in_sizes[0] / 8;
  const int threads_needed = (nrows + ROWS_PER_THREAD - 1) / ROWS_PER_THREAD;
  const int blocks = (threads_needed + BLOCK_SIZE - 1) / BLOCK_SIZE;
  const int total_threads = blocks * BLOCK_SIZE;

  cgp_kernel<<<blocks, BLOCK_SIZE, 0, stream>>>(
      (const v4f*)x, ephs, (v4f*)out, nrows, total_threads);
}